// Miner_45835890982944
// MI455X (gfx1250) — compile-verified
//
#include <hip/hip_runtime.h>
#include <hip/hip_bf16.h>
#include <math.h>

typedef __attribute__((ext_vector_type(2))) float v2f;
typedef __attribute__((ext_vector_type(8))) float v8f;

constexpr int N       = 8192;
constexpr int D       = 128;
constexpr int TILE    = 16;
constexpr int WAVES   = 8;                    // waves per mining block
constexpr int JT      = N / TILE;             // 512 column tiles
constexpr int JT_PW   = JT / WAVES;           // 64 tiles per wave (contiguous, ascending j)
constexpr int KSTEPS  = D / 4;                // 32 wmma_f32_16x16x4 steps over K=128
constexpr float POS_INF = __builtin_huge_valf();

// ---------------------------------------------------------------------------
// Kernel 1: row-wise L2 normalization. One wave per row (32 lanes x 4 floats).
// EN (4 MB) written to workspace; it stays resident in the 192 MB L2.
// ---------------------------------------------------------------------------
__global__ __launch_bounds__(256) void normalize_rows(const float* __restrict__ x,
                                                      float* __restrict__ en) {
  const int lane = threadIdx.x & 31;
  const int row  = blockIdx.x * (blockDim.x >> 5) + (threadIdx.x >> 5);
  const float4 v = *reinterpret_cast<const float4*>(x + row * D + lane * 4);
  float ss = v.x * v.x + v.y * v.y + v.z * v.z + v.w * v.w;
#pragma unroll
  for (int m = 1; m < 32; m <<= 1) ss += __shfl_xor(ss, m, 32);
  const float inv = 1.0f / fmaxf(sqrtf(ss), 1e-8f);
  float4 o;
  o.x = v.x * inv; o.y = v.y * inv; o.z = v.z * inv; o.w = v.w * inv;
  *reinterpret_cast<float4*>(en + row * D + lane * 4) = o;
}

// ---------------------------------------------------------------------------
// Kernel 2: fused Gram-tile GEMM (fp32 WMMA 16x16x4) + masked hardest-pos /
// hardest-neg row reduction. Never materializes the 8192x8192 matrix.
// Block b owns anchor rows [16b, 16b+16); wave w owns column stripe
// [w*1024, (w+1)*1024). A-fragment for full K held in registers.
// ---------------------------------------------------------------------------
__global__ __launch_bounds__(256) void mine_hardest(const float* __restrict__ en,
                                                    const int*   __restrict__ labels,
                                                    float*       __restrict__ out) {
  __shared__ float s_pv[WAVES][TILE];
  __shared__ float s_nv[WAVES][TILE];
  __shared__ int   s_pj[WAVES][TILE];
  __shared__ int   s_nj[WAVES][TILE];

  const int lane = threadIdx.x & 31;
  const int wave = threadIdx.x >> 5;
  const int half = lane >> 4;      // which 16-lane half
  const int l16  = lane & 15;
  const int i0   = blockIdx.x * TILE;
  const int kb   = 2 * half;       // K sub-offset per A/B fp32 WMMA layout

  // A fragment: afrag[s][v] = EN[i0 + l16][4s + 2*half + v]  (layout 16x4 f32 A)
  v2f afrag[KSTEPS];
  {
    const float* arow = en + (size_t)(i0 + l16) * D + kb;
#pragma unroll
    for (int s = 0; s < KSTEPS; ++s)
      afrag[s] = *reinterpret_cast<const v2f*>(arow + 4 * s);
  }

  // Labels of the 8 anchor rows this lane's C fragment covers.
  int labi[8];
#pragma unroll
  for (int v = 0; v < 8; ++v) labi[v] = labels[i0 + v + 8 * half];

  float posv[8], negv[8];
  int   posj[8], negj[8];
#pragma unroll
  for (int v = 0; v < 8; ++v) {
    posv[v] = -POS_INF; negv[v] = POS_INF;
    posj[v] = 0x7fffffff; negj[v] = 0x7fffffff;
  }

  const int jt_beg = wave * JT_PW;
  const int jt_end = jt_beg + JT_PW;
  for (int jt = jt_beg; jt < jt_end; ++jt) {
    const int j0 = jt * TILE;
    const int j  = j0 + l16;                 // this lane's column for all 8 C rows
    const int labj = labels[j];
    const float* brow = en + (size_t)j * D + kb;

    v8f c = {0.f, 0.f, 0.f, 0.f, 0.f, 0.f, 0.f, 0.f};
#pragma unroll
    for (int s = 0; s < KSTEPS; ++s) {
      const v2f b = *reinterpret_cast<const v2f*>(brow + 4 * s);
      c = __builtin_amdgcn_wmma_f32_16x16x4_f32(
          /*neg_a=*/false, afrag[s], /*neg_b=*/false, b,
          /*c_mod=*/(short)0, c, /*reuse_a=*/false, /*reuse_b=*/false);
    }

    // Epilogue: dist = 1 - dot; masked running max/min with first-occurrence
    // semantics (ascending j within a lane + strict compares).
#pragma unroll
    for (int v = 0; v < 8; ++v) {
      const float dist = 1.0f - c[v];
      const int   i    = i0 + v + 8 * half;
      const bool  same = (labi[v] == labj);
      if (same && (i != j)) {
        if (dist > posv[v]) { posv[v] = dist; posj[v] = j; }
      }
      if (!same) {
        if (dist < negv[v]) { negv[v] = dist; negj[v] = j; }
      }
    }
  }

  // Reduce across the 16 lanes holding each C row (width-16 butterfly keeps
  // the two halves independent). Ties -> lower column index (== first max).
#pragma unroll
  for (int m = 1; m < 16; m <<= 1) {
#pragma unroll
    for (int v = 0; v < 8; ++v) {
      float opv = __shfl_xor(posv[v], m, 16);
      int   opj = __shfl_xor(posj[v], m, 16);
      if (opv > posv[v] || (opv == posv[v] && opj < posj[v])) { posv[v] = opv; posj[v] = opj; }
      float onv = __shfl_xor(negv[v], m, 16);
      int   onj = __shfl_xor(negj[v], m, 16);
      if (onv < negv[v] || (onv == negv[v] && onj < negj[v])) { negv[v] = onv; negj[v] = onj; }
    }
  }

  if (l16 == 0) {
#pragma unroll
    for (int v = 0; v < 8; ++v) {
      const int r = v + 8 * half;
      s_pv[wave][r] = posv[v]; s_pj[wave][r] = posj[v];
      s_nv[wave][r] = negv[v]; s_nj[wave][r] = negj[v];
    }
  }
  __syncthreads();

  // Combine the 8 wave stripes (disjoint ascending j ranges) and emit.
  if (threadIdx.x < TILE) {
    const int r = threadIdx.x;
    const int i = i0 + r;
    float bpv = -POS_INF; int bpj = 0x7fffffff;
    float bnv =  POS_INF; int bnj = 0x7fffffff;
#pragma unroll
    for (int w = 0; w < WAVES; ++w) {
      const float pv = s_pv[w][r]; const int pj = s_pj[w][r];
      if (pv > bpv || (pv == bpv && pj < bpj)) { bpv = pv; bpj = pj; }
      const float nv = s_nv[w][r]; const int nj = s_nj[w][r];
      if (nv < bnv || (nv == bnv && nj < bnj)) { bnv = nv; bnj = nj; }
    }
    if (bpj == 0x7fffffff) bpj = 0;   // jnp.argmax of all -inf row -> 0
    if (bnj == 0x7fffffff) bnj = 0;

    out[3 * i + 0]  = (float)i;       // triplets: (anchor, hardest_pos, hardest_neg)
    out[3 * i + 1]  = (float)bpj;
    out[3 * i + 2]  = (float)bnj;
    out[3 * N + i]  = bpv;            // hardest_pos_dist
    out[4 * N + i]  = bnv;            // hardest_neg_dist
  }
}

// ---------------------------------------------------------------------------
extern "C" void kernel_launch(void* const* d_in, const int* in_sizes, int n_in,
                              void* d_out, int out_size, void* d_ws, size_t ws_size,
                              hipStream_t stream) {
  const float* emb    = (const float*)d_in[0];   // [8192,128] fp32
  const int*   labels = (const int*)d_in[1];     // [8192] int
  float*       out    = (float*)d_out;           // 8192*3 + 8192 + 8192
  float*       en     = (float*)d_ws;            // normalized embeddings, 4 MB

  (void)in_sizes; (void)n_in; (void)out_size; (void)ws_size;

  normalize_rows<<<N / 8, 256, 0, stream>>>(emb, en);   // 8 rows per block
  mine_hardest  <<<N / TILE, 256, 0, stream>>>(en, labels, out);
}